// OldEdgeFeaturesEmbeddingLayer_29008209117746
// MI455X (gfx1250) — compile-verified
//
#include <hip/hip_runtime.h>

#define BATCH 4
#define NNODE 512
#define NFEAT 16
#define EMBED 64
#define NEG_SLOPE 0.01f

typedef float v2f __attribute__((ext_vector_type(2)));
typedef float v4f __attribute__((ext_vector_type(4)));
typedef float v8f __attribute__((ext_vector_type(8)));

// One block per (b,i) row. 128 threads = 4 wave32; each wave handles 8 j-tiles
// of 16 rows (512 j total). Fuses: ef@w4^T -> LeakyReLU -> adj-weighted sum
// over j -> @w3^T, all in one pass over edge_features (64 MB stream).
__global__ __launch_bounds__(128) void edge_embed_kernel(
    const float* __restrict__ ef,   // (B,N,N,16)
    const float* __restrict__ adj,  // (B,N,N)
    const float* __restrict__ w4,   // (64,16)  x4 = ef @ w4^T
    const float* __restrict__ w3,   // (64,64)  out = summed @ w3^T
    float* __restrict__ out)        // (B,N,64)
{
  const int bi    = blockIdx.x;     // flat (b,i): 0..2047
  const int tid   = threadIdx.x;
  const int wave  = tid >> 5;
  const int lane  = tid & 31;
  const int laneM = lane & 15;
  const int hi    = lane >> 4;      // 0: lanes 0-15, 1: lanes 16-31

  __shared__ float partial[4][EMBED];
  __shared__ float summed[EMBED];

  // ---- preload B fragments of w4^T (K=16 features, N=64 embeds) ----
  // f32 16x16x4 B layout: VGPR0 holds (K=0 | K=2), VGPR1 holds (K=1 | K=3),
  // N = lane%16. B[k][n] = w4[n][k] (row-major (64,16)) -> per-lane float2.
  v2f bfrag[4][4];
#pragma unroll
  for (int nt = 0; nt < 4; ++nt)
#pragma unroll
    for (int kk = 0; kk < 4; ++kk)
      bfrag[nt][kk] = *(const v2f*)(w4 + (nt * 16 + laneM) * NFEAT + kk * 4 + 2 * hi);

  const float* efrow = ef  + (size_t)bi * NNODE * NFEAT;
  const float* adjr  = adj + (size_t)bi * NNODE;

  float sacc[4] = {0.f, 0.f, 0.f, 0.f};

#pragma unroll 2
  for (int t = 0; t < 8; ++t) {
    const int jbase = (wave * 8 + t) * 16;

    // A fragments: 16 j-rows x 16 features; A layout mirrors B (M = lane%16,
    // VGPR0 = K0|K2, VGPR1 = K1|K3) -> contiguous float2 per lane per k-step.
    v2f afrag[4];
#pragma unroll
    for (int kk = 0; kk < 4; ++kk)
      afrag[kk] = *(const v2f*)(efrow + (size_t)(jbase + laneM) * NFEAT + kk * 4 + 2 * hi);

    // adjacency weights: accumulator VGPR v in this lane maps to j = jbase + 8*hi + v
    v4f adv0 = *(const v4f*)(adjr + jbase + hi * 8);
    v4f adv1 = *(const v4f*)(adjr + jbase + hi * 8 + 4);
    float adv[8] = {adv0.x, adv0.y, adv0.z, adv0.w, adv1.x, adv1.y, adv1.z, adv1.w};

#pragma unroll
    for (int nt = 0; nt < 4; ++nt) {
      v8f c = {};
#pragma unroll
      for (int kk = 0; kk < 4; ++kk)
        c = __builtin_amdgcn_wmma_f32_16x16x4_f32(
                /*neg_a=*/false, afrag[kk],
                /*neg_b=*/false, bfrag[nt][kk],
                /*c_mod=*/(short)0, c,
                /*reuse_a=*/false, /*reuse_b=*/false);

      // LeakyReLU(x) == max(x, 0.01*x): mul + max (dual-issuable), no cmp/cndmask.
      // Then adj-weighted reduction over the 8 M(j) rows this lane holds.
      float s = 0.f;
#pragma unroll
      for (int v = 0; v < 8; ++v) {
        float x = c[v];
        float y = fmaxf(x, NEG_SLOPE * x);
        s = fmaf(adv[v], y, s);
      }
      sacc[nt] += s;
    }
  }

  // fold upper half-wave (M=8..15 partner rows share the same N column)
#pragma unroll
  for (int nt = 0; nt < 4; ++nt) {
    float s = sacc[nt] + __shfl_xor(sacc[nt], 16, 32);
    if (hi == 0) partial[wave][nt * 16 + laneM] = s;
  }
  __syncthreads();

  if (tid < EMBED)
    summed[tid] = partial[0][tid] + partial[1][tid] + partial[2][tid] + partial[3][tid];
  __syncthreads();

  // tiny epilogue GEMM: out[bi, o] = sum_e w3[o,e] * summed[e]
  if (tid < EMBED) {
    const float* w3r = w3 + tid * EMBED;
    float acc = 0.f;
#pragma unroll
    for (int e = 0; e < EMBED; ++e)
      acc = fmaf(w3r[e], summed[e], acc);
    out[(size_t)bi * EMBED + tid] = acc;
  }
}

extern "C" void kernel_launch(void* const* d_in, const int* in_sizes, int n_in,
                              void* d_out, int out_size, void* d_ws, size_t ws_size,
                              hipStream_t stream) {
  const float* ef  = (const float*)d_in[0];  // edge_features (4,512,512,16) f32
  const float* adj = (const float*)d_in[1];  // adj (4,512,512) f32
  const float* w4  = (const float*)d_in[2];  // (64,16) f32
  const float* w3  = (const float*)d_in[3];  // (64,64) f32
  float* out = (float*)d_out;                // (4,512,64) f32

  dim3 grid(BATCH * NNODE);  // 2048 blocks
  dim3 block(128);           // 4 wave32
  hipLaunchKernelGGL(edge_embed_kernel, grid, block, 0, stream, ef, adj, w4, w3, out);
}